// WNO1d_87497073755001
// MI455X (gfx1250) — compile-verified
//
#include <hip/hip_runtime.h>
#include <math.h>

typedef float v2f __attribute__((ext_vector_type(2)));
typedef float v8f __attribute__((ext_vector_type(8)));

#define WMMA_F32(a,b,c) __builtin_amdgcn_wmma_f32_16x16x4_f32(false,(a),false,(b),(short)0,(c),false,false)

// db6-style filters from the reference (ANA = REC_LO/REC_HI, SYN = DEC_LO/DEC_HI)
__constant__ float ANA_LO[12] = {
  0.11154074335008017f, 0.4946238903983854f, 0.7511339080215775f, 0.3152503517092432f,
  -0.22626469396516913f, -0.12976686756709563f, 0.09750160558707936f, 0.02752286553001629f,
  -0.031582039318031156f, 0.0005538422009938016f, 0.004777257511010651f, -0.00107730108499558f };
__constant__ float ANA_HI[12] = {
  -0.00107730108499558f, -0.004777257511010651f, 0.0005538422009938016f, 0.031582039318031156f,
  0.02752286553001629f, -0.09750160558707936f, -0.12976686756709563f, 0.22626469396516913f,
  0.3152503517092432f, -0.7511339080215775f, 0.4946238903983854f, -0.11154074335008017f };
__constant__ float SYN_LO[12] = {
  -0.00107730108499558f, 0.004777257511010651f, 0.0005538422009938016f, -0.031582039318031156f,
  0.02752286553001629f, 0.09750160558707936f, -0.12976686756709563f, -0.22626469396516913f,
  0.3152503517092432f, 0.7511339080215775f, 0.4946238903983854f, 0.11154074335008017f };
__constant__ float SYN_HI[12] = {
  -0.11154074335008017f, 0.4946238903983854f, -0.7511339080215775f, 0.3152503517092432f,
  0.22626469396516913f, -0.12976686756709563f, -0.09750160558707936f, 0.02752286553001629f,
  0.031582039318031156f, 0.0005538422009938016f, -0.004777257511010651f, -0.00107730108499558f };

__device__ __forceinline__ float gelu_exact(float x) {
  return 0.5f * x * (1.0f + erff(x * 0.70710678118654752440f));
}

// v[b][c][s] = x[b][s]*fc0w[0][c] + grid(s)*fc0w[1][c] + fc0b[c]
__global__ __launch_bounds__(256) void k_embed(const float* __restrict__ x,
                                               const float* __restrict__ fc0w,
                                               const float* __restrict__ fc0b,
                                               float* __restrict__ v) {
  int idx = blockIdx.x * 256 + threadIdx.x;          // == (b*64+c)*8192+s
  int s = idx & 8191;
  int c = (idx >> 13) & 63;
  int b = idx >> 19;
  float g = (float)s * (1.0f / 8191.0f);
  float xv = x[(b << 13) + s];
  v[idx] = xv * fc0w[c] + g * fc0w[64 + c] + fc0b[c];
}

// One DWT level: stride-2 correlation with symmetric padding.
__global__ __launch_bounds__(256) void k_dwt(const float* __restrict__ in,
                                             float* __restrict__ lo, float* __restrict__ hi,
                                             int n, int outlen, int left) {
  long idx = (long)blockIdx.x * 256 + threadIdx.x;
  long tot = 2048L * outlen;
  if (idx >= tot) return;
  int j = (int)(idx % outlen);
  long sig = idx / outlen;
  const float* src = in + sig * n;
  float aL = 0.f, aH = 0.f;
#pragma unroll
  for (int t = 0; t < 12; ++t) {
    int q = 2 * j + t - left;
    q = (q < 0) ? (-q - 1) : q;
    q = (q >= n) ? (2 * n - 1 - q) : q;
    float val = src[q];
    aL += val * ANA_LO[t];
    aH += val * ANA_HI[t];
  }
  lo[sig * outlen + j] = aL;
  hi[sig * outlen + j] = aH;
}

// One IDWT level: lhs_dilation=2, pad=1 each side, out = 2m-10.
__global__ __launch_bounds__(256) void k_idwt(const float* __restrict__ lo, int lostride,
                                              const float* __restrict__ hi, int m,
                                              float* __restrict__ out, int outn) {
  long idx = (long)blockIdx.x * 256 + threadIdx.x;
  long tot = 2048L * outn;
  if (idx >= tot) return;
  int j = (int)(idx % outn);
  long sig = idx / outn;
  const float* lp = lo + sig * lostride;
  const float* hp = hi + sig * (long)m;
  float acc = 0.f;
#pragma unroll
  for (int t = 0; t < 12; ++t) {
    int p = j + t - 1;                     // index into dilated (zeros-inserted) signal
    if (p >= 0 && (p & 1) == 0) {
      int u = p >> 1;
      if (u < m) acc += lp[u] * SYN_LO[t] + hp[u] * SYN_HI[t];
    }
  }
  out[sig * outn + j] = acc;
}

// Per mode x: out[b][o] = sum_i y[b][i] * w[i][o].  M=32, N=64, K=64. WMMA f32 16x16x4.
__global__ __launch_bounds__(256) void k_modemix(const float* __restrict__ y,
                                                 const float* __restrict__ w,
                                                 float* __restrict__ outy) {
  __shared__ float As[32 * 64];   // A[b][i]
  __shared__ float Bs[64 * 64];   // B[i][o]
  const int x = blockIdx.x;       // 0..41
  const int tid = threadIdx.x;
  for (int i = tid; i < 2048; i += 256) As[i] = y[(long)i * 42 + x];
  for (int i = tid; i < 4096; i += 256) Bs[i] = w[(long)i * 42 + x];
  __syncthreads();
  int wave = tid >> 5, lane = tid & 31;
  int half = lane >> 4, lm = lane & 15;
  int mt = (wave >> 2) * 16;      // 0/16 (batch tile)
  int nt = (wave & 3) * 16;       // 0..48 (out-chan tile)
  v8f acc = {};
#pragma unroll
  for (int k0 = 0; k0 < 64; k0 += 4) {
    int k = k0 + 2 * half;
    v2f a, b;
    a.x = As[(mt + lm) * 64 + k];     a.y = As[(mt + lm) * 64 + k + 1];
    b.x = Bs[k * 64 + nt + lm];       b.y = Bs[(k + 1) * 64 + nt + lm];
    acc = WMMA_F32(a, b, acc);
  }
#pragma unroll
  for (int r = 0; r < 8; ++r) {
    int bb = mt + r + half * 8;
    int oo = nt + lm;
    outy[((long)bb * 64 + oo) * 42 + x] = acc[r];
  }
}

// vout[b][o][s] = act( x1[b][o][s] + sum_i cw[i][o]*v[b][i][s] + cb[o] )
__global__ __launch_bounds__(256) void k_combine(const float* __restrict__ v,
                                                 const float* __restrict__ x1,
                                                 const float* __restrict__ cw,
                                                 const float* __restrict__ cb,
                                                 float* __restrict__ vout, int act) {
  __shared__ float As[64 * 64];   // A[o][i] = cw[i][o]
  __shared__ float Bs[64 * 32];   // B[i][n] = v[b][i][s0+n]
  const int s0 = blockIdx.x * 32;
  const int b = blockIdx.y;
  const int tid = threadIdx.x;
  for (int i = tid; i < 4096; i += 256) {
    int o = i >> 6, c = i & 63;
    As[i] = cw[c * 64 + o];
  }
  const long vbase = (long)b * 64 * 8192;
  for (int i = tid; i < 2048; i += 256) {
    int c = i >> 5, n = i & 31;
    Bs[i] = v[vbase + (long)c * 8192 + s0 + n];
  }
  __syncthreads();
  int wave = tid >> 5, lane = tid & 31;
  int half = lane >> 4, lm = lane & 15;
  int mt = (wave & 3) * 16;       // out-channel tile 0..48
  int nt = (wave >> 2) * 16;      // position tile 0/16
  v8f acc = {};
#pragma unroll
  for (int k0 = 0; k0 < 64; k0 += 4) {
    int k = k0 + 2 * half;
    v2f a, b2;
    a.x = As[(mt + lm) * 64 + k];   a.y = As[(mt + lm) * 64 + k + 1];
    b2.x = Bs[k * 32 + nt + lm];    b2.y = Bs[(k + 1) * 32 + nt + lm];
    acc = WMMA_F32(a, b2, acc);
  }
#pragma unroll
  for (int r = 0; r < 8; ++r) {
    int o = mt + r + half * 8;
    int s = s0 + nt + lm;
    long off = vbase + (long)o * 8192 + s;
    float val = acc[r] + x1[off] + cb[o];
    vout[off] = act ? gelu_exact(val) : val;
  }
}

// Head: h = gelu(v^T @ fc1_w + fc1_b) [32 x 128]; out = h @ fc2_w + fc2_b
__global__ __launch_bounds__(256) void k_head(const float* __restrict__ v,
                                              const float* __restrict__ w1,
                                              const float* __restrict__ b1,
                                              const float* __restrict__ w2,
                                              const float* __restrict__ b2,
                                              float* __restrict__ out) {
  __shared__ float As[32 * 64];     // A[m][c] (m = s-row)
  __shared__ float Bs[64 * 128];    // fc1_w
  __shared__ float Hs[32 * 128];    // gelu(fc1) staging
  const int s0 = blockIdx.x * 32;
  const int b = blockIdx.y;
  const int tid = threadIdx.x;
  const long vbase = (long)b * 64 * 8192;
  for (int i = tid; i < 2048; i += 256) {
    int c = i >> 5, m = i & 31;
    As[m * 64 + c] = v[vbase + (long)c * 8192 + s0 + m];
  }
  for (int i = tid; i < 8192; i += 256) Bs[i] = w1[i];
  __syncthreads();
  int wave = tid >> 5, lane = tid & 31;
  int half = lane >> 4, lm = lane & 15;
  int mt = (wave & 1) * 16;
  int ntb = (wave >> 1) * 16;       // 0..48; each wave also does ntb+64
#pragma unroll
  for (int rep = 0; rep < 2; ++rep) {
    int nt = ntb + rep * 64;
    v8f acc = {};
#pragma unroll
    for (int k0 = 0; k0 < 64; k0 += 4) {
      int k = k0 + 2 * half;
      v2f a, bb;
      a.x = As[(mt + lm) * 64 + k];   a.y = As[(mt + lm) * 64 + k + 1];
      bb.x = Bs[k * 128 + nt + lm];   bb.y = Bs[(k + 1) * 128 + nt + lm];
      acc = WMMA_F32(a, bb, acc);
    }
#pragma unroll
    for (int r = 0; r < 8; ++r) {
      int m = mt + r + half * 8;
      int j = nt + lm;
      Hs[m * 128 + j] = gelu_exact(acc[r] + b1[j]);
    }
  }
  __syncthreads();
  if (tid < 32) {                    // deterministic fc2 reduction
    float acc = 0.f;
#pragma unroll 8
    for (int j = 0; j < 128; ++j) acc += Hs[tid * 128 + j] * w2[j];
    out[(long)b * 8192 + s0 + tid] = acc + b2[0];
  }
}

extern "C" void kernel_launch(void* const* d_in, const int* in_sizes, int n_in,
                              void* d_out, int out_size, void* d_ws, size_t ws_size,
                              hipStream_t stream) {
  (void)in_sizes; (void)n_in; (void)out_size; (void)ws_size;
  const float* x    = (const float*)d_in[0];
  const float* fc0w = (const float*)d_in[1];
  const float* fc0b = (const float*)d_in[2];
  const float* w1   = (const float*)d_in[3];
  const float* w2   = (const float*)d_in[4];
  const float* cw   = (const float*)d_in[5];
  const float* cb   = (const float*)d_in[6];
  const float* fc1w = (const float*)d_in[7];
  const float* fc1b = (const float*)d_in[8];
  const float* fc2w = (const float*)d_in[9];
  const float* fc2b = (const float*)d_in[10];
  float* out = (float*)d_out;

  const int B = 32, C = 64, S = 8192, NL = 4;
  const int SIG = B * C;            // 2048 signals

  int lens[9]; lens[0] = S;
  int outl[8], lefts[8];
  for (int k = 0; k < 8; ++k) {
    int n = lens[k];
    int ol = (n + 11) / 2;
    outl[k] = ol; lens[k + 1] = ol;
    lefts[k] = (2 * (ol - 1) - n + 12) / 2;   // symmetric left pad (always 10)
  }
  long hiOff[8]; long hiTot = 0;
  for (int k = 0; k < 8; ++k) { hiOff[k] = hiTot; hiTot += (long)SIG * outl[k]; }
  const int modes = lens[8];        // 42

  const long VS = (long)SIG * S;
  float* base = (float*)d_ws;
  float* vA  = base;
  float* vB  = vA + VS;
  float* x1b = vB + VS;
  float* hib = x1b + VS;
  float* loA = hib + hiTot;                         // max 4102 per sig (idwt k=1)
  float* loB = loA + (long)SIG * (outl[0] + 2);
  float* ylm = loB + (long)SIG * outl[1];
  float* yhm = ylm + (long)SIG * modes;

  { // embed (fc0 + grid)
    long tot = (long)B * C * S;
    k_embed<<<(int)(tot / 256), 256, 0, stream>>>(x, fc0w, fc0b, vA);
  }

  float* vcur = vA; float* vnext = vB;
  float* lop[2] = { loA, loB };

  for (int l = 0; l < NL; ++l) {
    // ---- forward DWT pyramid ----
    const float* cur = vcur;
    for (int k = 0; k < 8; ++k) {
      float* lo = lop[k & 1];
      float* hi = hib + hiOff[k];
      long tot = (long)SIG * outl[k];
      k_dwt<<<(int)((tot + 255) / 256), 256, 0, stream>>>(cur, lo, hi, lens[k], outl[k], lefts[k]);
      cur = lo;
    }
    const float* yl = cur;          // level-8 lo, len 42 (lives in loB)

    // ---- mode mixing (WMMA) ----
    k_modemix<<<modes, 256, 0, stream>>>(yl, w1 + (long)l * C * C * modes, ylm);
    k_modemix<<<modes, 256, 0, stream>>>(hib + hiOff[7], w2 + (long)l * C * C * modes, yhm);

    // ---- inverse DWT ----
    const float* lo = ylm; int lostride = modes;
    for (int k = 7; k >= 0; --k) {
      const float* hi = (k == 7) ? yhm : (hib + hiOff[k]);
      int m = outl[k];              // truncate lo to h length via stride+m
      int outn = 2 * m - 10;
      float* dst = (k == 0) ? x1b : lop[(7 - k) & 1];
      long tot = (long)SIG * outn;
      k_idwt<<<(int)((tot + 255) / 256), 256, 0, stream>>>(lo, lostride, hi, m, dst, outn);
      lo = dst; lostride = outn;
    }

    // ---- pointwise channel GEMM + residual + gelu (WMMA) ----
    dim3 grid(S / 32, B);
    k_combine<<<grid, 256, 0, stream>>>(vcur, x1b, cw + (long)l * C * C, cb + (long)l * C,
                                        vnext, (l < NL - 1) ? 1 : 0);
    float* t = vcur; vcur = vnext; vnext = t;
  }

  // ---- head: fc1+gelu+fc2 fused (WMMA) ----
  dim3 g2(S / 32, B);
  k_head<<<g2, 256, 0, stream>>>(vcur, fc1w, fc1b, fc2w, fc2b, out);
}